// CourseGNN_27943057228202
// MI455X (gfx1250) — compile-verified
//
#include <hip/hip_runtime.h>
#include <hip/hip_bf16.h>

typedef __attribute__((ext_vector_type(2))) float v2f;
typedef __attribute__((ext_vector_type(8))) float v8f;
typedef unsigned int u32;
typedef u32 u32x4 __attribute__((ext_vector_type(4)));
typedef int i32x4 __attribute__((ext_vector_type(4)));
typedef int i32x8 __attribute__((ext_vector_type(8)));

#if defined(__has_builtin)
#  if __has_builtin(__builtin_amdgcn_tensor_load_to_lds)
#    define GCN_HAS_TDM 1
#  else
#    define GCN_HAS_TDM 0
#  endif
#else
#  define GCN_HAS_TDM 0
#endif

// ---------------------------------------------------------------------------
// Small utility kernels for CSR construction
// ---------------------------------------------------------------------------

__global__ void gcn_zero_i32(int* p, int n) {
    int i = blockIdx.x * blockDim.x + threadIdx.x;
    if (i < n) p[i] = 0;
}

// Count in-degree per destination node (self-loops added later as +1).
__global__ void gcn_count_kernel(const long long* __restrict__ ei, int E, int* __restrict__ cnt) {
    int e = blockIdx.x * blockDim.x + threadIdx.x;
    if (e < E) {
        int d = (int)ei[(size_t)E + e];   // edge_index[1][e]
        atomicAdd(&cnt[d], 1);
    }
}

// dinv[i] = rsqrt(deg) with deg = cnt + 1 (self loop) >= 1
__global__ void gcn_dinv_kernel(const int* __restrict__ cnt, float* __restrict__ dinv, int n) {
    int i = blockIdx.x * blockDim.x + threadIdx.x;
    if (i < n) dinv[i] = rsqrtf((float)(cnt[i] + 1));
}

// Single-workgroup exclusive scan over (cnt[i]+1) -> row_ptr[0..n]
__global__ void gcn_scan_kernel(const int* __restrict__ cnt, int* __restrict__ row_ptr, int n) {
    __shared__ int sdata[1024];
    __shared__ int carry;
    const int tid = threadIdx.x;
    if (tid == 0) carry = 0;
    for (int base = 0; base < n; base += 1024) {
        __syncthreads();
        int c = carry;
        int i = base + tid;
        int v = (i < n) ? (cnt[i] + 1) : 0;
        sdata[tid] = v;
        __syncthreads();
        #pragma unroll
        for (int off = 1; off < 1024; off <<= 1) {
            int t = (tid >= off) ? sdata[tid - off] : 0;
            __syncthreads();
            sdata[tid] += t;
            __syncthreads();
        }
        if (i < n) row_ptr[i] = c + sdata[tid] - v;   // exclusive
        __syncthreads();
        if (tid == 0) carry = c + sdata[1023];
    }
    __syncthreads();
    if (tid == 0) row_ptr[n] = carry;
}

// Place the self-loop entry first in each bucket, init fill cursor past it.
__global__ void gcn_selfloop_kernel(const int* __restrict__ row_ptr,
                                    const float* __restrict__ dinv,
                                    int* __restrict__ cursor,
                                    int* __restrict__ csr_src,
                                    float* __restrict__ csr_w, int n) {
    int i = blockIdx.x * blockDim.x + threadIdx.x;
    if (i < n) {
        int p = row_ptr[i];
        csr_src[p] = i;
        float di = dinv[i];
        csr_w[p] = di * di;
        cursor[i] = p + 1;
    }
}

__global__ void gcn_fill_kernel(const long long* __restrict__ ei, int E,
                                const float* __restrict__ dinv,
                                int* __restrict__ cursor,
                                int* __restrict__ csr_src,
                                float* __restrict__ csr_w) {
    int e = blockIdx.x * blockDim.x + threadIdx.x;
    if (e < E) {
        int s = (int)ei[e];
        int d = (int)ei[(size_t)E + e];
        int p = atomicAdd(&cursor[d], 1);
        csr_src[p] = s;
        csr_w[p]   = dinv[s] * dinv[d];
    }
}

// ---------------------------------------------------------------------------
// FP32 WMMA GEMM:  Y[M,N] = X[M,K] @ W[K,N]
// One block computes a 16-row strip; wave w computes the 16x16 tile at n0=16w.
// The 16xK A tile is a CONTIGUOUS 16*K*4-byte block of X (tile spans full K),
// so it is staged into LDS with a single 1-D Tensor Data Mover descriptor
// issued by wave 0 (tracked by TENSORcnt), instead of a cooperative VALU loop.
// M must be a multiple of 16 (50000 = 3125*16).
// ---------------------------------------------------------------------------
template <int K, int N>
__global__ __launch_bounds__(32 * (N / 16)) void gcn_gemm_wmma(
        const float* __restrict__ X, const float* __restrict__ W,
        float* __restrict__ Y) {
    __shared__ float As[16 * K];
    const int m0   = blockIdx.x * 16;
    const int wave = threadIdx.x >> 5;
    const int lane = threadIdx.x & 31;
    const int n0   = wave * 16;

#if GCN_HAS_TDM
    if (wave == 0) {
        constexpr u32 nelem = 16u * K;                  // f32 elements in tile
        const unsigned long long ga =
            (unsigned long long)(const void*)(X + (size_t)m0 * K);
        // D# group 0: count=1 | lds_addr | global_addr | type=2
        u32x4 g0;
        g0.x = 1u;                                      // count=1 (valid user D#)
        g0.y = (u32)(unsigned long long)(uintptr_t)As;  // LDS byte offset (aperture low bits)
        g0.z = (u32)ga;                                 // global_addr[31:0]
        g0.w = (u32)(ga >> 32) | (2u << 30);            // global_addr[56:32] | type=2
        // D# group 1: data_size=4B, tensor_dim0 = tile_dim0 = nelem, stride = nelem
        i32x8 g1;
        g1[0] = (int)(2u << 16);                        // data_size=2 -> 4 bytes
        g1[1] = (int)((nelem & 0xFFFFu) << 16);         // tensor_dim0[15:0]  (bits 63:48)
        g1[2] = (int)(nelem >> 16);                     // tensor_dim0[31:16] (bits 79:64)
        g1[3] = (int)((nelem & 0xFFFFu) << 16);         // tile_dim0 (bits 127:112)
        g1[4] = 0;                                      // tile_dim1=0, tile_dim2=0
        g1[5] = (int)nelem;                             // tensor_dim0_stride[31:0]
        g1[6] = 0;
        g1[7] = 0;
        i32x4 z4 = {0, 0, 0, 0};
#if __clang_major__ >= 23
        i32x8 z8 = {0, 0, 0, 0, 0, 0, 0, 0};
        __builtin_amdgcn_tensor_load_to_lds(g0, g1, z4, z4, z8, 0);
#else
        __builtin_amdgcn_tensor_load_to_lds(g0, g1, z4, z4, 0);
#endif
        __builtin_amdgcn_s_wait_tensorcnt(0);           // wave 0 blocks until DMA done
    }
    __syncthreads();                                    // releases the other waves
#else
    {   // Fallback: vectorized cooperative stage (b128 stream).
        const float4* Xv  = (const float4*)(X + (size_t)m0 * K);
        float4*       Asv = (float4*)As;
        for (int idx = threadIdx.x; idx < 16 * K / 4; idx += blockDim.x)
            Asv[idx] = Xv[idx];
        __syncthreads();
    }
#endif

    v8f acc = {};
    const int mrow  = lane & 15;           // A: row M, B/D: column offset
    const int khalf = (lane >> 4) * 2;     // K sub-offset per lane half (0 or 2)

    #pragma unroll 4
    for (int k0 = 0; k0 < K; k0 += 4) {
        v2f a, b;
        // A 16x4 layout: lanes 0-15 hold K={0,1}, lanes 16-31 hold K={2,3}
        a.x = As[mrow * K + k0 + khalf];
        a.y = As[mrow * K + k0 + khalf + 1];
        // B 4x16 layout: row striped across lanes within a VGPR
        b.x = W[(size_t)(k0 + khalf) * N + n0 + mrow];
        b.y = W[(size_t)(k0 + khalf + 1) * N + n0 + mrow];
        acc = __builtin_amdgcn_wmma_f32_16x16x4_f32(
            /*neg_a=*/false, a, /*neg_b=*/false, b,
            /*c_mod=*/(short)0, acc, /*reuse_a=*/false, /*reuse_b=*/false);
    }

    // D layout: VGPR r -> M = r + 8*(lane>=16), N = n0 + (lane&15)
    const int mhalf = (lane >> 4) * 8;
    #pragma unroll
    for (int r = 0; r < 8; ++r)
        Y[(size_t)(m0 + mhalf + r) * N + n0 + mrow] = acc[r];
}

// ---------------------------------------------------------------------------
// CSR gather aggregation: out[node] = sum_e w[e] * in[src[e]] + bias (+ReLU)
// One wave per node; each lane owns C/32 contiguous channels (b128/b64 loads).
// Gather source (<= 25.6 MB) is L2-resident on MI455X (192 MB L2).
// ---------------------------------------------------------------------------
template <int C, bool RELU>
__global__ __launch_bounds__(256) void gcn_aggregate(
        const float* __restrict__ Yin,
        const int* __restrict__ row_ptr,
        const int* __restrict__ csr_src,
        const float* __restrict__ csr_w,
        const float* __restrict__ bias,
        float* __restrict__ Yout, int M) {
    constexpr int VEC = C / 32;
    const int node = blockIdx.x * (blockDim.x >> 5) + (threadIdx.x >> 5);
    if (node >= M) return;
    const int lane = threadIdx.x & 31;
    const int ch0  = lane * VEC;

    float acc[VEC];
    #pragma unroll
    for (int v = 0; v < VEC; ++v) acc[v] = 0.0f;

    const int beg = row_ptr[node], end = row_ptr[node + 1];
    for (int e = beg; e < end; ++e) {
        int   s = csr_src[e];
        float w = csr_w[e];
        const float* row = Yin + (size_t)s * C + ch0;
        if (e + 1 < end)   // hint next gather row toward the caches
            __builtin_prefetch(Yin + (size_t)csr_src[e + 1] * C + ch0, 0, 1);
        #pragma unroll
        for (int v = 0; v < VEC; ++v)
            acc[v] = fmaf(w, row[v], acc[v]);
    }

    float* out = Yout + (size_t)node * C + ch0;
    #pragma unroll
    for (int v = 0; v < VEC; ++v) {
        float r = acc[v] + bias[ch0 + v];
        if (RELU) r = fmaxf(r, 0.0f);
        out[v] = r;
    }
}

// ---------------------------------------------------------------------------
// Launch
// ---------------------------------------------------------------------------
extern "C" void kernel_launch(void* const* d_in, const int* in_sizes, int n_in,
                              void* d_out, int out_size, void* d_ws, size_t ws_size,
                              hipStream_t stream) {
    constexpr int IN_C = 256, HID_C = 128, OUT_C = 64;
    const float*      x  = (const float*)d_in[0];
    const long long*  ei = (const long long*)d_in[1];
    const float*      W1 = (const float*)d_in[2];
    const float*      b1 = (const float*)d_in[3];
    const float*      W2 = (const float*)d_in[4];
    const float*      b2 = (const float*)d_in[5];
    float*            out = (float*)d_out;

    const int N = in_sizes[0] / IN_C;     // 50000
    const int E = in_sizes[1] / 2;        // 800000
    const int NE = N + E;                 // CSR entries incl. self loops

    // Carve workspace (256B aligned regions).
    char* p = (char*)d_ws;
    auto carve = [&](size_t bytes) {
        char* r = p;
        p += (bytes + 255) & ~(size_t)255;
        return (void*)r;
    };
    int*   cnt     = (int*)  carve((size_t)N * 4);
    int*   row_ptr = (int*)  carve((size_t)(N + 1) * 4);
    int*   cursor  = (int*)  carve((size_t)N * 4);
    float* dinv    = (float*)carve((size_t)N * 4);
    int*   csr_src = (int*)  carve((size_t)NE * 4);
    float* csr_w   = (float*)carve((size_t)NE * 4);
    float* XW      = (float*)carve((size_t)N * HID_C * 4);  // reused for layer-2 pre-agg
    float* H       = (float*)carve((size_t)N * HID_C * 4);

    const int T = 256;
    const int gN = (N + T - 1) / T;
    const int gE = (E + T - 1) / T;

    // --- CSR build (graph is the same for both layers) ---
    gcn_zero_i32<<<gN, T, 0, stream>>>(cnt, N);
    gcn_count_kernel<<<gE, T, 0, stream>>>(ei, E, cnt);
    gcn_dinv_kernel<<<gN, T, 0, stream>>>(cnt, dinv, N);
    gcn_scan_kernel<<<1, 1024, 0, stream>>>(cnt, row_ptr, N);
    gcn_selfloop_kernel<<<gN, T, 0, stream>>>(row_ptr, dinv, cursor, csr_src, csr_w, N);
    gcn_fill_kernel<<<gE, T, 0, stream>>>(ei, E, dinv, cursor, csr_src, csr_w);

    // --- Layer 1: H = relu( Agg(X @ W1) + b1 ) ---
    gcn_gemm_wmma<IN_C, HID_C><<<N / 16, 32 * (HID_C / 16), 0, stream>>>(x, W1, XW);
    {
        const int nodes_per_blk = 256 / 32;
        gcn_aggregate<HID_C, true><<<(N + nodes_per_blk - 1) / nodes_per_blk, 256, 0, stream>>>(
            XW, row_ptr, csr_src, csr_w, b1, H, N);
    }

    // --- Layer 2: out = Agg(H @ W2) + b2 ---
    gcn_gemm_wmma<HID_C, OUT_C><<<N / 16, 32 * (OUT_C / 16), 0, stream>>>(H, W2, XW);
    {
        const int nodes_per_blk = 256 / 32;
        gcn_aggregate<OUT_C, false><<<(N + nodes_per_blk - 1) / nodes_per_blk, 256, 0, stream>>>(
            XW, row_ptr, csr_src, csr_w, b2, out, N);
    }
}